// RipsH1_76596446757483
// MI455X (gfx1250) — compile-verified
//
#include <hip/hip_runtime.h>
#include <math.h>

// CDNA5 / gfx1250 pairwise-distance kernel (RipsH1 deaths + dgm1).
//
// out[p] = || points[idx[2p]] - points[idx[2p+1]] ||_2   (points are 8-dim f32)
//
// Memory-bound: ~32MB index stream (NT loads) + ~16MB output (NT stores);
// 2MB points table stays resident in the 192MB L2. The 8-wide dot product is
// computed with V_WMMA_F32_16X16X4_F32: 16 distances/wave; one VGPR pair is
// simultaneously a valid A (16x4 = D) and B (4x16 = D^T) per the ISA layouts,
// so C += A*A accumulates the Gram matrix whose diagonal is the squared norms.

typedef __attribute__((ext_vector_type(2))) float v2f;
typedef __attribute__((ext_vector_type(8))) float v8f;
typedef __attribute__((ext_vector_type(2))) int   v2i;

__global__ __launch_bounds__(256) void pair_dist_wmma_kernel(
    const int* __restrict__ pairs,   // 2*n int32 indices, pair p = (pairs[2p], pairs[2p+1])
    const float* __restrict__ pts,   // [65536 x 8] f32, row-major
    float* __restrict__ out,         // n f32 distances
    int n)
{
    const int lane = threadIdx.x & 31;
    const int wave = blockIdx.x * (blockDim.x >> 5) + (threadIdx.x >> 5);
    const int base = wave * 16;            // 16 distances per wave tile
    const int j    = lane & 15;            // distance slot within tile
    const int hi   = lane >> 4;            // 0: K=0,1 (and K=4,5); 1: K=2,3 (and K=6,7)

    if (base + 16 <= n) {
        // ---- full tile: WMMA path (wave-uniform branch, EXEC all ones) ----
        const int g = base + j;
        // NT: index stream is read once; don't rinse L2 (keep points resident)
        const v2i ip = __builtin_nontemporal_load((const v2i*)(pairs + 2 * g));
        const float* pa = pts + (size_t)ip.x * 8;
        const float* pb = pts + (size_t)ip.y * 8;

        const int c0 = 2 * hi;       // components {c0, c0+1}   -> K-group 0
        const int c1 = 4 + 2 * hi;   // components {c1, c1+1}   -> K-group 1
        const v2f alo = *(const v2f*)(pa + c0);
        const v2f blo = *(const v2f*)(pb + c0);
        const v2f ahi = *(const v2f*)(pa + c1);
        const v2f bhi = *(const v2f*)(pb + c1);

        v2f A0; A0.x = alo.x - blo.x; A0.y = alo.y - blo.y;   // d[K0], d[K0+1]
        v2f A1; A1.x = ahi.x - bhi.x; A1.y = ahi.y - bhi.y;   // d[K1], d[K1+1]

        // C[m][n] = sum_k d_m[k]*d_n[k]; diagonal = squared norms.
        v8f c = {0.f, 0.f, 0.f, 0.f, 0.f, 0.f, 0.f, 0.f};
        c = __builtin_amdgcn_wmma_f32_16x16x4_f32(false, A0, false, A0,
                                                  (short)0, c, false, false);
        c = __builtin_amdgcn_wmma_f32_16x16x4_f32(false, A1, false, A1,
                                                  (short)0, c, false, false);

        // Diagonal (i,i): i<8 -> lane i, VGPR i ; i>=8 -> lane i+16, VGPR i-8.
        const int k = lane & 7;
        float r = c[0];
        r = (k == 1) ? c[1] : r;
        r = (k == 2) ? c[2] : r;
        r = (k == 3) ? c[3] : r;
        r = (k == 4) ? c[4] : r;
        r = (k == 5) ? c[5] : r;
        r = (k == 6) ? c[6] : r;
        r = (k == 7) ? c[7] : r;

        const bool active = (lane < 8) || (lane >= 24);
        const int  di     = (lane < 8) ? lane : (lane - 16);
        if (active) {
            // raw V_SQRT_F32 (<=1 ULP) instead of the precise libm expansion
            __builtin_nontemporal_store(__builtin_amdgcn_sqrtf(r), out + base + di);
        }
    } else {
        // ---- ragged tail: plain VALU path, lanes 0..15 cover remaining ----
        const int g = base + j;
        if (hi == 0 && g < n) {
            const v2i ip = *(const v2i*)(pairs + 2 * g);
            const float* pa = pts + (size_t)ip.x * 8;
            const float* pb = pts + (size_t)ip.y * 8;
            float acc = 0.f;
#pragma unroll
            for (int d = 0; d < 8; ++d) {
                const float t = pa[d] - pb[d];
                acc = fmaf(t, t, acc);
            }
            out[g] = __builtin_amdgcn_sqrtf(acc);
        }
    }
}

extern "C" void kernel_launch(void* const* d_in, const int* in_sizes, int n_in,
                              void* d_out, int out_size, void* d_ws, size_t ws_size,
                              hipStream_t stream) {
    const float* pts = (const float*)d_in[0];   // points  [65536 x 8] f32
    const int*   v0  = (const int*)d_in[1];     // verts0  flat int (65535*2)
    const int*   v1  = (const int*)d_in[2];     // verts1  flat int (2e6*4)
    float*       out = (float*)d_out;           // [deaths | dgm1] concatenated

    const int n0 = in_sizes[1] / 2;             // 65535 pairs
    const int n1 = in_sizes[2] / 2;             // 4,000,000 pairs

    // 256 threads = 8 wave32 = 128 distances per block
    const int blk = 256;
    const int per_block = (blk / 32) * 16;

    pair_dist_wmma_kernel<<<(n0 + per_block - 1) / per_block, blk, 0, stream>>>(
        v0, pts, out, n0);
    pair_dist_wmma_kernel<<<(n1 + per_block - 1) / per_block, blk, 0, stream>>>(
        v1, pts, out + n0, n1);
}